// TemporalGNN_472446403013
// MI455X (gfx1250) — compile-verified
//
#include <hip/hip_runtime.h>
#include <hip/hip_bf16.h>
#include <math.h>

// ---------------------------------------------------------------------------
// TemporalGNN (Dir-GCN x2 -> LSTM -> proj) for MI455X / gfx1250.
// fp32 WMMA (V_WMMA_F32_16X16X4_F32) + Tensor Data Mover staging of A-tiles.
// B=1, T=12, N=10000, F=64, H=128, E=160000. All GEMM dims multiples of 16.
// ---------------------------------------------------------------------------

typedef __attribute__((ext_vector_type(2))) float v2f;
typedef __attribute__((ext_vector_type(8))) float v8f;
typedef __attribute__((ext_vector_type(4))) unsigned int u32x4;
typedef __attribute__((ext_vector_type(8))) int i32x8;
typedef __attribute__((ext_vector_type(4))) int i32x4;

namespace {
constexpr int Tn  = 12;
constexpr int Nn  = 10000;
constexpr int Fin = 64;
constexpr int Hd  = 128;
constexpr int En  = 160000;
constexpr int G4H = 4 * Hd;  // 512
constexpr int LDS_PAD = 4;   // DWORDs of pad per row (TDM pad_amount = 4)
}

// ----------------------------- utility kernels -----------------------------

__global__ void tg_zero_kernel(float* __restrict__ p, int n) {
  int i = blockIdx.x * blockDim.x + threadIdx.x;
  if (i < n) p[i] = 0.0f;
}

__global__ void tg_degree_kernel(const int* __restrict__ src, const int* __restrict__ dst,
                                 float* __restrict__ dout, float* __restrict__ din, int E) {
  int e = blockIdx.x * blockDim.x + threadIdx.x;
  if (e < E) {
    atomicAdd(&dout[src[e]], 1.0f);
    atomicAdd(&din[dst[e]], 1.0f);
  }
}

__global__ void tg_coef_kernel(const int* __restrict__ src, const int* __restrict__ dst,
                               const float* __restrict__ dout, const float* __restrict__ din,
                               float* __restrict__ coef, int E) {
  int e = blockIdx.x * blockDim.x + threadIdx.x;
  if (e < E) {
    float a = dout[src[e]];
    float b = din[dst[e]];
    float ia = (a > 0.0f) ? rsqrtf(a) : 0.0f;
    float ib = (b > 0.0f) ? rsqrtf(b) : 0.0f;
    coef[e] = ia * ib;
  }
}

// Bidirectional message scatter: aggF[dst] += x[src]*coef, aggB[src] += x[dst]*coef.
__global__ void tg_scatter_kernel(const float* __restrict__ x, const int* __restrict__ src,
                                  const int* __restrict__ dst, const float* __restrict__ coef,
                                  float* __restrict__ aggF, float* __restrict__ aggB,
                                  int E, int Fc) {
  const int F4 = Fc >> 2;
  long tid = (long)blockIdx.x * blockDim.x + threadIdx.x;
  if (tid >= (long)E * F4) return;
  int e = (int)(tid / F4);
  int f = (int)(tid - (long)e * F4) << 2;
  int s = src[e], d = dst[e];
  float c = coef[e];
  const float4 xs = *reinterpret_cast<const float4*>(x + (size_t)s * Fc + f);
  const float4 xd = *reinterpret_cast<const float4*>(x + (size_t)d * Fc + f);
  float* pf = aggF + (size_t)d * Fc + f;
  float* pb = aggB + (size_t)s * Fc + f;
  atomicAdd(pf + 0, xs.x * c); atomicAdd(pf + 1, xs.y * c);
  atomicAdd(pf + 2, xs.z * c); atomicAdd(pf + 3, xs.w * c);
  atomicAdd(pb + 0, xd.x * c); atomicAdd(pb + 1, xd.y * c);
  atomicAdd(pb + 2, xd.z * c); atomicAdd(pb + 3, xd.w * c);
}

__global__ void tg_bias_sum_kernel(const float* __restrict__ a, const float* __restrict__ b,
                                   float* __restrict__ o, int n) {
  int i = blockIdx.x * blockDim.x + threadIdx.x;
  if (i < n) o[i] = a[i] + b[i];
}

// ------------------------ TDM descriptor construction ----------------------
// Loads a 16-row x K-col fp32 tile (row stride K elements in memory) into LDS
// at lds_byte_addr, padding LDS_PAD DWORDs after every K DWORDs so the tile
// lands with row stride (K + LDS_PAD) -> bank-conflict-free column reads.
// Descriptor layout per CDNA5 ISA (D# groups 0/1).
__device__ __forceinline__ void tg_tdm_load_tile(const float* gptr, unsigned lds_byte_addr,
                                                 int K) {
  const unsigned long long ga = (unsigned long long)(uintptr_t)gptr;
  // pad_interval code: 0=2,1=4,...,7=256 DWORDs => code = log2(K) - 1.
  const unsigned pi = (K == 64) ? 5u : 6u;

  u32x4 g0;
  g0.x = 0x1u;                                   // count=1 (valid), no gather
  g0.y = lds_byte_addr;                          // lds_addr[31:0]
  g0.z = (unsigned)(ga & 0xFFFFFFFFull);         // global_addr[31:0]
  g0.w = (unsigned)((ga >> 32) & 0x01FFFFFFull)  // global_addr[56:32]
         | (2u << 30);                           // type=2 ("image")

  i32x8 g1;
  g1[0] = (int)((2u << 16)                       // data_size=2 -> 4 bytes
                | (1u << 20)                     // pad_enable
                | (pi << 22)                     // pad_interval (K DWORDs)
                | ((unsigned)(LDS_PAD - 1) << 25));  // pad_amount: code 3 = 4 DWORDs
  g1[1] = (int)(((unsigned)K & 0xFFFFu) << 16);  // tensor_dim0[15:0] = K
  g1[2] = (int)(16u << 16);                      // tensor_dim0 hi=0 | tensor_dim1 lo = 16
  g1[3] = (int)(((unsigned)K & 0xFFFFu) << 16);  // tensor_dim1 hi=0 | tile_dim0 = K
  g1[4] = 16;                                    // tile_dim1 = 16, tile_dim2 = 0
  g1[5] = K;                                     // tensor_dim0_stride[31:0] = K
  g1[6] = 0;                                     // stride hi | tensor_dim1_stride lo
  g1[7] = 0;

  const i32x4 z4 = {0, 0, 0, 0};                 // groups 2/3 unused (2D tensor)
  const i32x8 z8 = {0, 0, 0, 0, 0, 0, 0, 0};
  __builtin_amdgcn_tensor_load_to_lds(g0, g1, z4, z4, z8, 0);
}

// ----------------------------- WMMA dual GEMM ------------------------------
// G1 = A1 @ W1^T, optionally G2 = A2 @ W2^T, fp32 WMMA. W is (Nout, K) row-major.
//   MODE 0 (GCN):   out = relu(alpha*(G1 + b1) + (1-alpha)*(G2 + b2))
//   MODE 1 (gates): out = G1 + G2 + b1          (b1 = bih + bhh, precombined)
//   MODE 2 (proj):  out = G1 + b1
// Block = 256 threads (8 waves) -> one 16-row x 128-col output slab.
// A-tiles staged by the Tensor Data Mover (one DMA per matrix, issued by wave 0,
// completion via TENSORcnt), deposited with padded row stride K+4.
template <int MODE>
__global__ __launch_bounds__(256)
void tg_wmma_gemm_kernel(const float* __restrict__ A1, const float* __restrict__ W1,
                         const float* __restrict__ b1,
                         const float* __restrict__ A2, const float* __restrict__ W2,
                         const float* __restrict__ b2,
                         const float* __restrict__ alpha_p,
                         float* __restrict__ out, int M, int K, int Nout) {
  __shared__ float As[2 * 16 * (128 + LDS_PAD)];  // max footprint (K=128)

  const int tid  = threadIdx.x;
  const int lane = tid & 31;
  const int wave = tid >> 5;
  const int m0   = blockIdx.x * 16;
  const int lds_stride = K + LDS_PAD;             // 68 (K=64) / 132 (K=128)
  const int tile1_off  = 16 * lds_stride;         // second A tile, in floats

  // --- TDM stage: wave 0 issues the descriptor DMA(s), waits on TENSORcnt ---
  if (wave == 0) {
    const unsigned lds_base = (unsigned)(uintptr_t)(void*)&As[0];
    tg_tdm_load_tile(A1 + (size_t)m0 * K, lds_base, K);
    if (MODE != 2)
      tg_tdm_load_tile(A2 + (size_t)m0 * K, lds_base + (unsigned)tile1_off * 4u, K);
    __builtin_amdgcn_s_wait_tensorcnt(0);
  }
  __syncthreads();

  const int n0 = (blockIdx.y * 8 + wave) * 16;
  if (n0 >= Nout) return;  // wave-uniform: EXEC stays all-ones for WMMA

  // f32 16x16x4 fragment indexing (ISA 7.12.2):
  //   A (16x4): lane 0-15 -> M=lane, K={0,1}; lane 16-31 -> M=lane-16, K={2,3}
  //   B (4x16): lane 0-15 -> N=lane, K={0,1}; lane 16-31 -> N=lane-16, K={2,3}
  //   C (16x16): VGPR v, lane l -> row v + 8*(l>>4), col l&15
  const int mrow = lane & 15;
  const int koff = (lane >> 4) << 1;
  const int ncol = n0 + (lane & 15);

  const float* a1row = &As[mrow * lds_stride];
  const float* a2row = &As[tile1_off + mrow * lds_stride];
  const float* w1row = W1 + (size_t)ncol * K;
  const float* w2row = (MODE != 2) ? (W2 + (size_t)ncol * K) : nullptr;

  v8f acc1 = {};
  v8f acc2 = {};
  for (int k = 0; k < K; k += 4) {
    const int kb = k + koff;
    v2f a1; a1.x = a1row[kb]; a1.y = a1row[kb + 1];
    v2f p1; p1.x = w1row[kb]; p1.y = w1row[kb + 1];
    acc1 = __builtin_amdgcn_wmma_f32_16x16x4_f32(false, a1, false, p1,
                                                 (short)0, acc1, false, false);
    if (MODE != 2) {
      v2f a2; a2.x = a2row[kb]; a2.y = a2row[kb + 1];
      v2f p2; p2.x = w2row[kb]; p2.y = w2row[kb + 1];
      acc2 = __builtin_amdgcn_wmma_f32_16x16x4_f32(false, a2, false, p2,
                                                   (short)0, acc2, false, false);
    }
  }

  const float al    = (MODE == 0) ? alpha_p[0] : 0.0f;
  const float bias1 = b1 ? b1[ncol] : 0.0f;
  const float bias2 = (MODE == 0) ? b2[ncol] : 0.0f;

#pragma unroll
  for (int v = 0; v < 8; ++v) {
    const int m = m0 + v + ((lane >> 4) << 3);
    float val;
    if (MODE == 0) {
      val = al * (acc1[v] + bias1) + (1.0f - al) * (acc2[v] + bias2);
      val = val > 0.0f ? val : 0.0f;  // relu
    } else if (MODE == 1) {
      val = acc1[v] + acc2[v] + bias1;
    } else {
      val = acc1[v] + bias1;
    }
    out[(size_t)m * Nout + ncol] = val;
  }
}

// ------------------------------- LSTM cell ---------------------------------

__global__ void tg_lstm_cell_kernel(const float* __restrict__ gates,
                                    float* __restrict__ h, float* __restrict__ c,
                                    int S, int H) {
  int i = blockIdx.x * blockDim.x + threadIdx.x;
  if (i >= S * H) return;
  int s = i / H;
  int j = i - s * H;
  const float* g = gates + (size_t)s * 4 * H;
  float ig = g[j];
  float fg = g[H + j];
  float gg = g[2 * H + j];
  float og = g[3 * H + j];
  float si = 1.0f / (1.0f + __expf(-ig));
  float sf = 1.0f / (1.0f + __expf(-fg));
  float so = 1.0f / (1.0f + __expf(-og));
  float cn = sf * c[i] + si * tanhf(gg);
  c[i] = cn;
  h[i] = so * tanhf(cn);
}

// ------------------------------- host driver -------------------------------

static inline void tg_zero(float* p, size_t n, hipStream_t s) {
  tg_zero_kernel<<<dim3((unsigned)((n + 255) / 256)), dim3(256), 0, s>>>(p, (int)n);
}

extern "C" void kernel_launch(void* const* d_in, const int* in_sizes, int n_in,
                              void* d_out, int out_size, void* d_ws, size_t ws_size,
                              hipStream_t stream) {
  (void)in_sizes; (void)n_in; (void)out_size; (void)ws_size;

  const float* x_seq = (const float*)d_in[0];   // (1, T, N, F)
  const int*   ei    = (const int*)d_in[1];     // (T, 2, E)
  const float* gw1_0 = (const float*)d_in[2];   // (H, F)
  const float* gb1_0 = (const float*)d_in[3];   // (H)
  const float* gw2_0 = (const float*)d_in[4];
  const float* gb2_0 = (const float*)d_in[5];
  const float* gw1_1 = (const float*)d_in[6];   // (H, H)
  const float* gb1_1 = (const float*)d_in[7];
  const float* gw2_1 = (const float*)d_in[8];
  const float* gb2_1 = (const float*)d_in[9];
  const float* wih   = (const float*)d_in[10];  // (4H, H)
  const float* whh   = (const float*)d_in[11];  // (4H, H)
  const float* bih   = (const float*)d_in[12];  // (4H)
  const float* bhh   = (const float*)d_in[13];  // (4H)
  const float* wp    = (const float*)d_in[14];  // (F, H)
  const float* bp    = (const float*)d_in[15];  // (F)
  const float* alpha = (const float*)d_in[16];  // (1)
  float* out = (float*)d_out;                   // (1, N, F)

  // Workspace carve-up (floats, 256B-aligned chunks). Total ~108 MB.
  float* ws = (float*)d_ws;
  size_t off = 0;
  auto alloc = [&](size_t n) {
    float* p = ws + off;
    off += (n + 63) & ~(size_t)63;
    return p;
  };
  float* degO  = alloc(Nn);
  float* degI  = alloc(Nn);
  float* coef  = alloc(En);
  float* aggF  = alloc((size_t)Nn * Hd);
  float* aggB  = alloc((size_t)Nn * Hd);
  float* h1    = alloc((size_t)Nn * Hd);
  float* gnn   = alloc((size_t)Tn * Nn * Hd);  // (T, N, H)
  float* gates = alloc((size_t)Nn * G4H);
  float* hb    = alloc((size_t)Nn * Hd);
  float* cb    = alloc((size_t)Nn * Hd);
  float* bsum  = alloc(G4H);

  const dim3 blk256(256);
  const dim3 gridE((En + 255) / 256);
  const unsigned mtiles = Nn / 16;  // 625

  // Combined LSTM bias (bih + bhh), used once per recurrent step.
  tg_bias_sum_kernel<<<dim3((G4H + 255) / 256), blk256, 0, stream>>>(bih, bhh, bsum, G4H);

  // ---------------- GNN over timesteps ----------------
  for (int t = 0; t < Tn; ++t) {
    const int* src = ei + (size_t)t * 2 * En;
    const int* dst = src + En;
    const float* x_t = x_seq + (size_t)t * Nn * Fin;

    tg_zero(degO, Nn, stream);
    tg_zero(degI, Nn, stream);
    tg_degree_kernel<<<gridE, blk256, 0, stream>>>(src, dst, degO, degI, En);
    tg_coef_kernel<<<gridE, blk256, 0, stream>>>(src, dst, degO, degI, coef, En);

    // Layer 0: F=64 -> H=128
    tg_zero(aggF, (size_t)Nn * Fin, stream);
    tg_zero(aggB, (size_t)Nn * Fin, stream);
    {
      long total = (long)En * (Fin / 4);
      tg_scatter_kernel<<<dim3((unsigned)((total + 255) / 256)), blk256, 0, stream>>>(
          x_t, src, dst, coef, aggF, aggB, En, Fin);
    }
    tg_wmma_gemm_kernel<0><<<dim3(mtiles, 1), blk256, 0, stream>>>(
        aggF, gw1_0, gb1_0, aggB, gw2_0, gb2_0, alpha, h1, Nn, Fin, Hd);

    // Layer 1: H=128 -> H=128
    tg_zero(aggF, (size_t)Nn * Hd, stream);
    tg_zero(aggB, (size_t)Nn * Hd, stream);
    {
      long total = (long)En * (Hd / 4);
      tg_scatter_kernel<<<dim3((unsigned)((total + 255) / 256)), blk256, 0, stream>>>(
          h1, src, dst, coef, aggF, aggB, En, Hd);
    }
    tg_wmma_gemm_kernel<0><<<dim3(mtiles, 1), blk256, 0, stream>>>(
        aggF, gw1_1, gb1_1, aggB, gw2_1, gb2_1, alpha,
        gnn + (size_t)t * Nn * Hd, Nn, Hd, Hd);
  }

  // ---------------- LSTM recurrence ----------------
  tg_zero(hb, (size_t)Nn * Hd, stream);
  tg_zero(cb, (size_t)Nn * Hd, stream);
  for (int t = 0; t < Tn; ++t) {
    // gates = x_t @ wih^T + h @ whh^T + (bih + bhh), fused dual-GEMM.
    tg_wmma_gemm_kernel<1><<<dim3(mtiles, G4H / 128), blk256, 0, stream>>>(
        gnn + (size_t)t * Nn * Hd, wih, bsum, hb, whh, nullptr, nullptr,
        gates, Nn, Hd, G4H);
    tg_lstm_cell_kernel<<<dim3((Nn * Hd + 255) / 256), blk256, 0, stream>>>(
        gates, hb, cb, Nn, Hd);
  }

  // ---------------- Output projection ----------------
  tg_wmma_gemm_kernel<2><<<dim3(mtiles, 1), blk256, 0, stream>>>(
      hb, wp, bp, nullptr, nullptr, nullptr, nullptr, out, Nn, Hd, Fin);
}